// Signature_75101798137928
// MI455X (gfx1250) — compile-verified
//
#include <hip/hip_runtime.h>

// CDNA5 / gfx1250, wave32. Depth-4 path signature via Chen's identity.
// Level-4 update per step expressed as a 144x144 += (144x28)@(28x144) GEMM
// executed with V_WMMA_F32_16X16X4_F32; sig4 held in per-wave WMMA
// accumulator registers for the whole 127-step scan.

typedef __attribute__((ext_vector_type(2))) float v2f;
typedef __attribute__((ext_vector_type(8))) float v8f;

#define NC    12              // channels
#define C2    144             // 12^2
#define C3    1728            // 12^3
#define C4    20736           // 12^4
#define LPTS  128             // stream length
#define NINC  127             // increments
#define KP    28              // K = 25 padded to 28 (7 x k4 chunks)
#define NTHR  288             // 9 waves of 32
#define SIGSZ (NC + C2 + C3 + C4)  // 22620

__launch_bounds__(NTHR, 1)
__global__ void sig_depth4_kernel(const float* __restrict__ path,
                                  float* __restrict__ out) {
  __shared__ float dx[NINC * NC];                 // all increments, 6096 B
  __shared__ float s1[NC];
  __shared__ float s2[C2];
  __shared__ float s3[C3];
  __shared__ __align__(16) float Am[C2 * KP];     // A, row-major 144x28
  __shared__ __align__(16) float Bt[C2 * KP];     // B^T, row-major 144x28

  const int b    = blockIdx.x;
  const int tid  = threadIdx.x;
  const int lane = tid & 31;
  const int wv   = tid >> 5;                      // 0..8 (tile row)
  const float* p = path + (size_t)b * LPTS * NC;

  // ---- preload path increments into LDS ----
  for (int idx = tid; idx < NINC * NC; idx += NTHR) {
    int t = idx / NC, c = idx - t * NC;
    dx[idx] = p[(t + 1) * NC + c] - p[t * NC + c];
  }
  // ---- zero state (running Chen product starting from the zero signature
  //      reproduces init = exp_terms(dx0) after the first step) ----
  if (tid < NC) s1[tid] = 0.f;
  if (tid < C2) s2[tid] = 0.f;
  for (int idx = tid; idx < C3; idx += NTHR) s3[idx] = 0.f;

  v8f acc[9];
#pragma unroll
  for (int j = 0; j < 9; ++j)
#pragma unroll
    for (int v = 0; v < 8; ++v) acc[j][v] = 0.f;

  const int m  = lane & 15;                       // M (and N) sub-index
  const int kb = (lane >> 4) << 1;                // K base: 0 or 2

  float n1 = 0.f, n2 = 0.f, n3r[6];

  for (int t = 0; t < NINC; ++t) {
    __syncthreads();  // (1) prev s-writes visible; prev A/Bt reads retired
    const float* d = &dx[t * NC];

    // ---- build A = [sig3 | sig1 (x) I | sig2 + e2/6 | 0-pad], 14 elems/thread
#pragma unroll
    for (int r = 0; r < 14; ++r) {
      int idx = tid + r * NTHR;                   // 0..4031
      int ij = idx / KP, c = idx - ij * KP;
      int i = ij / 12, j = ij - i * 12;
      float v;
      if (c < 12)       v = s3[ij * 12 + c];
      else if (c < 24)  v = (j == (c - 12)) ? s1[i] : 0.f;
      else if (c == 24) v = s2[ij] + d[i] * d[j] * (1.f / 12.f);
      else              v = 0.f;
      Am[idx] = v;
    }
    // ---- build B^T = [I (x) d^T ; e3 ; e2^T ; 0-pad]^T, 14 elems/thread
#pragma unroll
    for (int r = 0; r < 14; ++r) {
      int idx = tid + r * NTHR;
      int kl = idx / KP, c = idx - kl * KP;
      int k = kl / 12, l = kl - k * 12;
      float v;
      if (c < 12)       v = (c == k) ? d[l] : 0.f;
      else if (c < 24)  v = d[c - 12] * d[k] * d[l] * (1.f / 6.f);
      else if (c == 24) v = d[k] * d[l] * 0.5f;
      else              v = 0.f;
      Bt[idx] = v;
    }
    // ---- levels <=3 update (reads OLD s1/s2/s3) into registers
    if (tid < NC) n1 = s1[tid] + d[tid];
    if (tid < C2) {
      int i = tid / 12, j = tid - i * 12;
      n2 = s2[tid] + s1[i] * d[j] + d[i] * d[j] * 0.5f;
    }
#pragma unroll
    for (int r = 0; r < 6; ++r) {
      int idx = tid + r * NTHR;                   // 0..1727
      int i = idx / 144, jk = idx - i * 144;
      int j = jk / 12, k = jk - j * 12;
      n3r[r] = s3[idx] + s2[i * 12 + j] * d[k]
             + s1[i] * d[j] * d[k] * 0.5f
             + d[i] * d[j] * d[k] * (1.f / 6.f);
    }
    __syncthreads();  // (2) A/Bt ready; old-s reads retired

    // ---- write back small levels (no reader until next barrier (1))
    if (tid < NC) s1[tid] = n1;
    if (tid < C2) s2[tid] = n2;
#pragma unroll
    for (int r = 0; r < 6; ++r) s3[tid + r * NTHR] = n3r[r];

    // ---- level-4: sig4 += A @ B via 9x7 fp32 WMMAs per wave.
    // Uniform control flow: EXEC is all-1s here as WMMA requires.
    v2f a[7];
#pragma unroll
    for (int kc = 0; kc < 7; ++kc)
      a[kc] = *(const v2f*)&Am[(16 * wv + m) * KP + 4 * kc + kb];
#pragma unroll
    for (int j = 0; j < 9; ++j) {
#pragma unroll
      for (int kc = 0; kc < 7; ++kc) {
        v2f bf = *(const v2f*)&Bt[(16 * j + m) * KP + 4 * kc + kb];
        acc[j] = __builtin_amdgcn_wmma_f32_16x16x4_f32(
            /*neg_a=*/false, a[kc], /*neg_b=*/false, bf,
            /*c_mod=*/(short)0, acc[j], /*reuse_a=*/false, /*reuse_b=*/false);
      }
    }
  }

  __syncthreads();

  // ---- store: levels 1..3 from LDS, level 4 straight from accumulators
  float* ob = out + (size_t)b * SIGSZ;
  if (tid < NC) ob[tid] = s1[tid];
  if (tid < C2) ob[NC + tid] = s2[tid];
#pragma unroll
  for (int r = 0; r < 6; ++r) {
    int idx = tid + r * NTHR;
    ob[NC + C2 + idx] = s3[idx];
  }
  // C/D layout: vgpr v, lanes 0-15 -> (row=v,   col=lane),
  //                     lanes 16-31 -> (row=v+8, col=lane-16)
  float* o4 = ob + NC + C2 + C3;
  const int rowoff = (lane >= 16) ? 8 : 0;
#pragma unroll
  for (int j = 0; j < 9; ++j)
#pragma unroll
    for (int v = 0; v < 8; ++v) {
      int row = 16 * wv + v + rowoff;
      o4[row * 144 + 16 * j + m] = acc[j][v];
    }
}

extern "C" void kernel_launch(void* const* d_in, const int* in_sizes, int n_in,
                              void* d_out, int out_size, void* d_ws, size_t ws_size,
                              hipStream_t stream) {
  const float* path = (const float*)d_in[0];
  float* out = (float*)d_out;
  (void)in_sizes; (void)n_in; (void)out_size; (void)d_ws; (void)ws_size;
  sig_depth4_kernel<<<dim3(128), dim3(NTHR), 0, stream>>>(path, out);
}